// SpatialInSARSignalModel_85779086835971
// MI455X (gfx1250) — compile-verified
//
#include <hip/hip_runtime.h>
#include <hip/hip_bf16.h>
#include <math.h>

typedef __attribute__((ext_vector_type(2))) float v2f;
typedef __attribute__((ext_vector_type(8))) float v8f;

#define N_TIME     365
#define T_PAD      368      // 23 tiles of 16
#define K_PAD      12       // rank 10 padded to 12 (3 x K=4 WMMA steps)
#define N_SLICES   3        // K_PAD / 4
#define SMOOTH     0.1f
#define TWO_PI     6.28318530717958647692f

// ---------------------------------------------------------------------------
// Kernel 1: build A (N x 12):  [c, m, amp*cos(ph) x4, amp*sin(ph) x4, 0, 0]
// amp/ph are the 5-neighbor smoothed values.
// ---------------------------------------------------------------------------
__global__ void prep_A_kernel(const float* __restrict__ c_off,
                              const float* __restrict__ trend,
                              const float* __restrict__ amps,     // N x 4
                              const float* __restrict__ phs,      // N x 4
                              const float* __restrict__ wgt,      // N x 5
                              const int*   __restrict__ idx,      // N x 5
                              float* __restrict__ A,              // N x 12
                              int n_stations) {
    int n = blockIdx.x * blockDim.x + threadIdx.x;
    if (n >= n_stations) return;

    float wa[4] = {0.f, 0.f, 0.f, 0.f};
    float wp[4] = {0.f, 0.f, 0.f, 0.f};
    #pragma unroll
    for (int j = 0; j < 5; ++j) {
        int   nb = idx[n * 5 + j];
        float w  = wgt[n * 5 + j];
        #pragma unroll
        for (int f = 0; f < 4; ++f) {
            wa[f] = fmaf(w, amps[nb * 4 + f], wa[f]);
            wp[f] = fmaf(w, phs [nb * 4 + f], wp[f]);
        }
    }

    float* Ar = A + (size_t)n * K_PAD;
    Ar[0] = c_off[n];
    Ar[1] = trend[n];
    #pragma unroll
    for (int f = 0; f < 4; ++f) {
        float a = (1.0f - SMOOTH) * amps[n * 4 + f] + SMOOTH * wa[f];
        float p = (1.0f - SMOOTH) * phs [n * 4 + f] + SMOOTH * wp[f];
        float sp, cp;
        __sincosf(p, &sp, &cp);
        Ar[2 + f] = a * cp;   // pairs with B row 2+f = sin(w_f * t)
        Ar[6 + f] = a * sp;   // pairs with B row 6+f = cos(w_f * t)
    }
    Ar[10] = 0.0f;
    Ar[11] = 0.0f;
}

// ---------------------------------------------------------------------------
// Kernel 2: build B pre-swizzled into WMMA 4x16-f32 operand order.
// Logical rows: [1, t, sin(w_f t) x4, cos(w_f t) x4, 0, 0].
// Packed layout: Bp[(slice*T_PAD + col)*4 + k], slice = row>>2, k = row&3,
// so a lane's fragment {k=kb, k=kb+1} is one contiguous 8B load.
// ---------------------------------------------------------------------------
__global__ void prep_B_kernel(const float* __restrict__ time_vec, // 365
                              const float* __restrict__ periods,  // 4
                              float* __restrict__ Bp) {           // 3*368*4
    int j = blockIdx.x * blockDim.x + threadIdx.x;
    if (j >= T_PAD) return;
    float t = (j < N_TIME) ? time_vec[j] : 0.0f;

    float row[K_PAD];
    row[0] = 1.0f;
    row[1] = t;
    #pragma unroll
    for (int f = 0; f < 4; ++f) {
        float theta = TWO_PI * (1.0f / periods[f]) * t;
        float s, c;
        __sincosf(theta, &s, &c);
        row[2 + f] = s;
        row[6 + f] = c;
    }
    row[10] = 0.0f;
    row[11] = 0.0f;

    #pragma unroll
    for (int r = 0; r < K_PAD; ++r) {
        int s = r >> 2, k = r & 3;
        Bp[(s * T_PAD + j) * 4 + k] = row[r];
    }
}

// ---------------------------------------------------------------------------
// Kernel 3: out(N x 365) = A(N x 12) * B(12 x 368), fp32 WMMA 16x16x4.
// 8 waves / block; each wave owns one 16-row strip, sweeps 23 column tiles.
// Pre-swizzled B staged in LDS (17.7 KB): one ds_load_b64 per B fragment,
// bank-conflict-free. Output written with non-temporal stores (streaming,
// write-once, never re-read -> keep it out of L2's way).
// EXEC stays all-ones (wave-uniform guards) around the WMMAs.
// ---------------------------------------------------------------------------
#define WAVES_PER_BLOCK 8

__global__ void wmma_gemm_kernel(const float* __restrict__ A,   // N x 12
                                 const float* __restrict__ Bp,  // 3*368*4 packed
                                 float* __restrict__ out,       // N x 365
                                 int n_stations) {
    __shared__ float Bs[N_SLICES * T_PAD * 4];   // 17664 B

    for (int i = threadIdx.x; i < N_SLICES * T_PAD * 4; i += blockDim.x)
        Bs[i] = Bp[i];
    __syncthreads();

    const int wave  = threadIdx.x >> 5;
    const int lane  = threadIdx.x & 31;
    const int mtile = blockIdx.x * WAVES_PER_BLOCK + wave;   // wave-uniform
    const int n_mtiles = n_stations >> 4;                    // N % 16 == 0
    if (mtile >= n_mtiles) return;                           // whole-wave exit

    const int lo16 = lane & 15;
    const int kb   = (lane >> 4) << 1;      // lanes 0-15 -> K{0,1}; 16-31 -> K{2,3}
    const int baseRow = mtile << 4;

    // A operand: 16x4 f32 tile per K-slice, 2 VGPRs/lane.
    const float* Arow = A + (size_t)(baseRow + lo16) * K_PAD + kb;
    v2f a0, a1, a2;
    a0.x = Arow[0]; a0.y = Arow[1];   // slice 0: K = kb,  kb+1
    a1.x = Arow[4]; a1.y = Arow[5];   // slice 1: K = 4+kb, 5+kb
    a2.x = Arow[8]; a2.y = Arow[9];   // slice 2: K = 8+kb, 9+kb

    const int rowBase = baseRow + ((lane >> 4) << 3);  // + v gives M of acc[v]

    for (int ct = 0; ct < T_PAD / 16; ++ct) {
        const int col = (ct << 4) + lo16;

        // One contiguous, 8B-aligned b64 LDS load per B fragment.
        v2f b0 = *(const v2f*)&Bs[(0 * T_PAD + col) * 4 + kb];
        v2f b1 = *(const v2f*)&Bs[(1 * T_PAD + col) * 4 + kb];
        v2f b2 = *(const v2f*)&Bs[(2 * T_PAD + col) * 4 + kb];

        v8f acc = {};
        acc = __builtin_amdgcn_wmma_f32_16x16x4_f32(false, a0, false, b0,
                                                    (short)0, acc, false, false);
        acc = __builtin_amdgcn_wmma_f32_16x16x4_f32(false, a1, false, b1,
                                                    (short)0, acc, false, false);
        acc = __builtin_amdgcn_wmma_f32_16x16x4_f32(false, a2, false, b2,
                                                    (short)0, acc, false, false);

        if (col < N_TIME) {
            #pragma unroll
            for (int v = 0; v < 8; ++v)
                __builtin_nontemporal_store(acc[v],
                    &out[(size_t)(rowBase + v) * N_TIME + col]);
        }
    }
}

// ---------------------------------------------------------------------------
extern "C" void kernel_launch(void* const* d_in, const int* in_sizes, int n_in,
                              void* d_out, int out_size, void* d_ws, size_t ws_size,
                              hipStream_t stream) {
    const float* time_vec = (const float*)d_in[0];   // 365
    const float* c_off    = (const float*)d_in[1];   // N
    const float* trend    = (const float*)d_in[2];   // N
    const float* amps     = (const float*)d_in[3];   // N x 4
    const float* phs      = (const float*)d_in[4];   // N x 4
    const float* wgt      = (const float*)d_in[5];   // N x 5
    const float* periods  = (const float*)d_in[6];   // 4
    const int*   idx      = (const int*)  d_in[7];   // N x 5
    float* out = (float*)d_out;

    const int N = in_sizes[1];

    // Workspace layout: packed B at offset 0 (17.7 KB), A at 32 KB (4.8 MB).
    float* Bp = (float*)d_ws;
    float* A  = (float*)((char*)d_ws + 32768);

    hipLaunchKernelGGL(prep_A_kernel, dim3((N + 255) / 256), dim3(256), 0, stream,
                       c_off, trend, amps, phs, wgt, idx, A, N);

    hipLaunchKernelGGL(prep_B_kernel, dim3((T_PAD + 127) / 128), dim3(128), 0, stream,
                       time_vec, periods, Bp);

    const int n_mtiles = N / 16;
    const int grid = (n_mtiles + WAVES_PER_BLOCK - 1) / WAVES_PER_BLOCK;
    hipLaunchKernelGGL(wmma_gemm_kernel, dim3(grid), dim3(WAVES_PER_BLOCK * 32), 0, stream,
                       A, Bp, out, N);
}